// LearnablePatchify_14405320311383
// MI455X (gfx1250) — compile-verified
//
#include <hip/hip_runtime.h>

typedef __attribute__((ext_vector_type(2))) float v2f;
typedef __attribute__((ext_vector_type(8))) float v8f;
typedef __attribute__((ext_vector_type(4))) unsigned int v4u;
typedef __attribute__((ext_vector_type(8))) int v8i;
typedef __attribute__((ext_vector_type(4))) int v4i;

#define PATCH  28
#define KTOT   2352            // 3 * 28 * 28
#define IMG    448
#define CB     3
#define NPATCH 256             // 16 * 16 patches per image
#define PP     784             // 28 * 28 output elements per patch
#define CROWS  14              // image rows per TDM chunk
#define CELEMS (CROWS * IMG)   // 6272 floats = 25088 B, contiguous in memory
#define NCHUNK 6               // 3 channels * 2 half-slabs

// Issue one TDM descriptor: contiguous 25KB strip (14 image rows of the
// 16-patch band for channel c) -> LDS. 1-D tile: tile_dim0 = 6272 elems,
// data_size = 4B. EXEC is ignored by TENSOR_LOAD_TO_LDS, so this must only
// be reached by one wave (wave-uniform branch in the caller).
__device__ __forceinline__ void tdm_load_chunk(const float* x, int b, int hp,
                                               int q, unsigned int lds_off) {
    const int c  = q >> 1;
    const int i0 = (q & 1) * CROWS;
    const unsigned long long gaddr =
        (unsigned long long)(uintptr_t)x +
        (unsigned long long)((b * CB + c) * IMG + hp * PATCH + i0) * IMG * 4ull;

    v4u g0;
    g0.x = 1u;                                    // count=1 (valid user D#)
    g0.y = lds_off;                               // lds_addr (byte offset)
    g0.z = (unsigned int)gaddr;                   // global_addr[31:0]
    g0.w = (unsigned int)((gaddr >> 32) & 0x01FFFFFFull) | 0x80000000u; // [56:32] | type=2

    v8i g1;
    g1[0] = 0x00020000;                           // data_size=2 (4B); no mask/flags
    g1[1] = (int)((CELEMS & 0xFFFF) << 16);       // tensor_dim0[15:0]
    g1[2] = (int)((CELEMS >> 16) | (1 << 16));    // tensor_dim0[31:16] | tensor_dim1=1
    g1[3] = (int)(CELEMS << 16);                  // tensor_dim1 hi | tile_dim0=6272
    g1[4] = 0;                                    // tile_dim1=0, tile_dim2=0 (1-D)
    g1[5] = CELEMS;                               // tensor_dim0_stride[31:0]
    g1[6] = 0;                                    // stride0 hi | stride1 lo
    g1[7] = 0;                                    // stride1 hi
    const v4i z4 = {0, 0, 0, 0};
    const v8i z8 = {0, 0, 0, 0, 0, 0, 0, 0};
    // 6-arg form (clang-23 / therock-10.0 headers): extra descriptor group
    // (unused for a 2-group 1-D descriptor) + cache policy.
    __builtin_amdgcn_tensor_load_to_lds(g0, g1, z4, z4, z8, 0);
}

// One block = one patch-row (b, hp): 16 patches along wp = WMMA M dim.
// V_WMMA_F32_16X16X4_F32 with A from LDS (TDM-filled), B = channel-mean
// weights broadcast to all 16 columns -> every D column holds the patch dot
// product (exact fp32 FMA semantics).
__global__ __launch_bounds__(128) void patchify_tdm_wmma_kernel(
    const float* __restrict__ x,      // [64,3,448,448]
    const float* __restrict__ W,      // [3,3,28,28]
    const float* __restrict__ bias,   // [3]
    float* __restrict__ out)          // [64,256,28,28]
{
    __shared__ float buf[2][CELEMS];      // 50,176 B double buffer
    __shared__ float wm[KTOT];            //  9,408 B channel-mean weights
    __shared__ float partial[4][16];      // per-wave patch partial sums
    __shared__ float pres[16];            // final patch values

    const int tid  = threadIdx.x;
    const int wave = tid >> 5;
    const int lane = tid & 31;
    const int b    = blockIdx.x >> 4;
    const int hp   = blockIdx.x & 15;

    const unsigned int boff[2] = {
        (unsigned int)(uintptr_t)&buf[0][0],
        (unsigned int)(uintptr_t)&buf[1][0]
    };

    // Kick off chunk 0 DMA immediately (overlaps with weight prep).
    if (tid < 32) {
        tdm_load_chunk(x, b, hp, 0, boff[0]);
    }

    // Channel-mean of conv weights -> LDS (mean over out-channels commutes
    // with the einsum + channel-mean in the reference).
    for (int k = tid; k < KTOT; k += 128) {
        wm[k] = (W[k] + W[k + KTOT] + W[k + 2 * KTOT]) * (1.0f / 3.0f);
    }
    const float bmean = (bias[0] + bias[1] + bias[2]) * (1.0f / 3.0f);

    const int wp   = lane & 15;        // WMMA M index (patch along the row)
    const int ksel = (lane >> 4) << 1; // lanes<16 -> K=0,1 ; lanes>=16 -> K=2,3

    v8f acc0 = {};
    v8f acc1 = {};

    for (int q = 0; q < NCHUNK; ++q) {
        if (tid < 32) {
            if (q < NCHUNK - 1) {
                tdm_load_chunk(x, b, hp, q + 1, boff[(q + 1) & 1]);
                __builtin_amdgcn_s_wait_tensorcnt((short)1);  // chunk q done
            } else {
                __builtin_amdgcn_s_wait_tensorcnt((short)0);
            }
        }
        __syncthreads();   // buffer q visible to all waves (also covers wm on q==0)

        const float* s  = &buf[q & 1][0];
        const int    c  = q >> 1;
        const int    i0 = (q & 1) * CROWS;
        // Waves split the 14 rows of this chunk round-robin.
        for (int r = wave; r < CROWS; r += 4) {
            const float* arow = s + r * IMG + wp * PATCH + ksel;
            const float* brow = &wm[c * PP + (i0 + r) * PATCH + ksel];
#pragma unroll
            for (int jb = 0; jb < PATCH; jb += 8) {
                {
                    const float2 av = *reinterpret_cast<const float2*>(arow + jb);
                    const float2 bv = *reinterpret_cast<const float2*>(brow + jb);
                    v2f a;  a.x  = av.x; a.y  = av.y;
                    v2f w2; w2.x = bv.x; w2.y = bv.y;
                    acc0 = __builtin_amdgcn_wmma_f32_16x16x4_f32(
                        false, a, false, w2, (short)0, acc0, false, false);
                }
                if (jb + 4 < PATCH) {
                    const float2 av = *reinterpret_cast<const float2*>(arow + jb + 4);
                    const float2 bv = *reinterpret_cast<const float2*>(brow + jb + 4);
                    v2f a;  a.x  = av.x; a.y  = av.y;
                    v2f w2; w2.x = bv.x; w2.y = bv.y;
                    acc1 = __builtin_amdgcn_wmma_f32_16x16x4_f32(
                        false, a, false, w2, (short)0, acc1, false, false);
                }
            }
        }
        __syncthreads();   // all waves done with buffer (q&1) before its reuse
    }

    // D layout: VGPR g, lanes 0-15 = D[M=g][N=lane], lanes 16-31 = D[M=8+g].
    // All columns are equal, so lanes 0 and 16 publish the 16 patch partials.
    const v8f accs = acc0 + acc1;
    if (lane == 0) {
#pragma unroll
        for (int g = 0; g < 8; ++g) partial[wave][g] = accs[g];
    }
    if (lane == 16) {
#pragma unroll
        for (int g = 0; g < 8; ++g) partial[wave][8 + g] = accs[g];
    }
    __syncthreads();
    if (tid < 16) {
        pres[tid] = partial[0][tid] + partial[1][tid] +
                    partial[2][tid] + partial[3][tid] + bmean;
    }
    __syncthreads();

    // Broadcast epilogue: 16 patches x 784 floats, coalesced float4 stores.
    float* outp = out + (size_t)(b * NPATCH + hp * 16) * PP;
#pragma unroll
    for (int m = 0; m < 16; ++m) {
        const float v = pres[m];
        const float4 vv = make_float4(v, v, v, v);
        float4* po = reinterpret_cast<float4*>(outp + m * PP);
        for (int t = tid; t < PP / 4; t += 128) {
            po[t] = vv;
        }
    }
}

extern "C" void kernel_launch(void* const* d_in, const int* in_sizes, int n_in,
                              void* d_out, int out_size, void* d_ws, size_t ws_size,
                              hipStream_t stream) {
    (void)in_sizes; (void)n_in; (void)d_ws; (void)ws_size; (void)out_size;
    const float* x    = (const float*)d_in[0];
    const float* W    = (const float*)d_in[1];
    const float* bias = (const float*)d_in[2];
    // d_in[3] = patch_size scalar (28), baked into kernel constants.
    float* out = (float*)d_out;

    // 1024 patch-rows -> one block (4 waves) each.
    patchify_tdm_wmma_kernel<<<dim3(1024), dim3(128), 0, stream>>>(x, W, bias, out);
}